// S6Block_81836306858718
// MI455X (gfx1250) — compile-verified
//
#include <hip/hip_runtime.h>
#include <hip/hip_bf16.h>

// ---------------------------------------------------------------------------
// S6 (Mamba) block forward for MI455X (gfx1250), wave32 + WMMA f32 16x16x4.
//   B=2, S=2048, Dm=768, Di=1536, Ds=16, K=4
// K1 rmsnorm -> xn ; K2 WMMA GEMM -> xz ; K3 conv+silu -> xc ;
// K4 GEMV33 -> ssm ; K5 selective scan -> yb ; K6 WMMA GEMM (+res) -> out
// ---------------------------------------------------------------------------

typedef float v2f __attribute__((ext_vector_type(2)));
typedef float v8f __attribute__((ext_vector_type(8)));
typedef int   i32x4 __attribute__((vector_size(16)));   // matches async-LDS builtin pointee

#define BSZ   2
#define SEQ   2048
#define DM    768
#define DI    1536
#define DS    16
#define ROWS  (BSZ * SEQ)        // 4096

// ---- GEMM tiling: block 128x256 (8 waves as 2x4), wave tile 64x64 ----
#define BM 128
#define BN 256
#define BK 32
#define ASTR 36                  // floats per A row in LDS (144B, 16B-aligned, conflict-free)
#define BSTR 260                 // floats per B row in LDS (1040B, 16B-aligned)
#define A_BUF (BM * ASTR)        // 4608 floats
#define B_BUF (BK * BSTR)        // 8320 floats
#define SMEM_FLOATS (2 * (A_BUF + B_BUF))   // 25856 floats = 103424 bytes
#define ASYNC_OPS_PER_THREAD 12  // 4 (A) + 8 (B) b128 copies

#if defined(__has_builtin)
#if __has_builtin(__builtin_amdgcn_global_load_async_to_lds_b128) && \
    __has_builtin(__builtin_amdgcn_s_wait_asynccnt)
#define HAVE_ASYNC_LDS 1
#endif
#endif

__device__ __forceinline__ void cp16_to_lds(const float* g, float* l) {
#if defined(HAVE_ASYNC_LDS)
  __builtin_amdgcn_global_load_async_to_lds_b128(
      (__attribute__((address_space(1))) i32x4*)(uintptr_t)g,
      (__attribute__((address_space(3))) i32x4*)(uintptr_t)l,
      0, 0);
#else
  *(float4*)l = *(const float4*)g;
#endif
}

// ---------------- K1: RMSNorm ----------------
__global__ __launch_bounds__(256) void rmsnorm_k(const float* __restrict__ x,
                                                 const float* __restrict__ w,
                                                 float* __restrict__ xn) {
  const int row = blockIdx.x;
  const float* xr = x + (size_t)row * DM;
  float v[3];
  float s = 0.f;
#pragma unroll
  for (int i = 0; i < 3; ++i) {
    v[i] = xr[threadIdx.x + i * 256];
    s += v[i] * v[i];
  }
#pragma unroll
  for (int m = 16; m >= 1; m >>= 1) s += __shfl_xor(s, m, 32);
  __shared__ float red[8];
  if ((threadIdx.x & 31) == 0) red[threadIdx.x >> 5] = s;
  __syncthreads();
  float total = 0.f;
#pragma unroll
  for (int i = 0; i < 8; ++i) total += red[i];
  const float r = rsqrtf(total / (float)DM + 1e-6f);
  float* o = xn + (size_t)row * DM;
#pragma unroll
  for (int i = 0; i < 3; ++i)
    o[threadIdx.x + i * 256] = v[i] * r * w[threadIdx.x + i * 256];
}

// ---------------- K2/K6: fp32 WMMA GEMM, C = A@B (+ addend) ----------------
// A: MxK row-major, B: KxN row-major, C: MxN row-major.
// Double-buffered LDS fed by async global->LDS B128 copies (CDNA5 path).
__global__ __launch_bounds__(256) void gemm_wmma_f32(const float* __restrict__ A,
                                                     const float* __restrict__ B,
                                                     const float* __restrict__ addend,
                                                     float* __restrict__ C,
                                                     int M, int N, int K) {
  extern __shared__ float smem[];   // [2][A_BUF + B_BUF]

  const int tid  = threadIdx.x;
  const int lane = tid & 31;
  const int wave = tid >> 5;   // 0..7
  const int wm   = wave >> 2;  // 0..1
  const int wn   = wave & 3;   // 0..3
  const int bm   = blockIdx.y * BM;
  const int bn   = blockIdx.x * BN;

  const int ml = lane & 15;
  const int kh = (lane >> 4) << 1;   // 0 (lanes 0-15) / 2 (lanes 16-31)

  v8f acc[4][4] = {};

  auto load_tiles = [&](int buf, int k0) {
    float* As = smem + buf * (A_BUF + B_BUF);
    float* Bs = As + A_BUF;
    // A tile: BM x BK = 4096 floats = 1024 x b128 -> 4 per thread.
#pragma unroll
    for (int i = 0; i < 4; ++i) {
      const int f = tid + i * 256;
      const int r = f >> 3, c4 = (f & 7) << 2;
      cp16_to_lds(A + (size_t)(bm + r) * K + k0 + c4, As + r * ASTR + c4);
    }
    // B tile: BK x BN = 8192 floats = 2048 x b128 -> 8 per thread.
#pragma unroll
    for (int i = 0; i < 8; ++i) {
      const int f = tid + i * 256;
      const int r = f >> 6, c4 = (f & 63) << 2;
      cp16_to_lds(B + (size_t)(k0 + r) * N + bn + c4, Bs + r * BSTR + c4);
    }
  };

  const int nk = K / BK;
  load_tiles(0, 0);

  for (int it = 0; it < nk; ++it) {
    const int cur = it & 1;
    if (it + 1 < nk) {
      load_tiles(cur ^ 1, (it + 1) * BK);
#if defined(HAVE_ASYNC_LDS)
      __builtin_amdgcn_s_wait_asynccnt(ASYNC_OPS_PER_THREAD);  // cur done; next in flight
#endif
    } else {
#if defined(HAVE_ASYNC_LDS)
      __builtin_amdgcn_s_wait_asynccnt(0);
#endif
    }
    __syncthreads();   // cur tiles visible to all waves

    const float* Asb = smem + cur * (A_BUF + B_BUF);
    const float* Bsb = Asb + A_BUF;

#pragma unroll
    for (int kk = 0; kk < BK; kk += 4) {
      v2f af[4], bf[4];
#pragma unroll
      for (int im = 0; im < 4; ++im) {
        // A 16x4 fragment: lane m = lane%16; lanes>=16 carry K+2/K+3 (one b64 read).
        const float* p = Asb + (size_t)(wm * 64 + im * 16 + ml) * ASTR + kk + kh;
        af[im][0] = p[0];
        af[im][1] = p[1];
      }
#pragma unroll
      for (int in = 0; in < 4; ++in) {
        // B 4x16 fragment (mirrored layout): lane n = lane%16.
        const float* p = Bsb + (size_t)(kk + kh) * BSTR + wn * 64 + in * 16 + ml;
        bf[in][0] = p[0];
        bf[in][1] = p[BSTR];
      }
#pragma unroll
      for (int im = 0; im < 4; ++im)
#pragma unroll
        for (int in = 0; in < 4; ++in)
          acc[im][in] = __builtin_amdgcn_wmma_f32_16x16x4_f32(
              false, af[im], false, bf[in], (short)0, acc[im][in], false, false);
    }
    __syncthreads();   // all reads of cur done before it is refilled
  }

  // Epilogue: C/D layout -> VGPR i holds row (i + 8*(lane/16)), col lane%16.
  const int mh = (lane >> 4) * 8;
#pragma unroll
  for (int im = 0; im < 4; ++im)
#pragma unroll
    for (int in = 0; in < 4; ++in) {
      const int mbase = bm + wm * 64 + im * 16 + mh;
      const int n     = bn + wn * 64 + in * 16 + ml;
#pragma unroll
      for (int i = 0; i < 8; ++i) {
        const size_t off = (size_t)(mbase + i) * N + n;
        float v = acc[im][in][i];
        if (addend) v += addend[off];
        C[off] = v;
      }
    }
}

// ---------------- K3: depthwise causal conv(K=4) + SiLU ----------------
__global__ __launch_bounds__(256) void conv_silu_k(const float* __restrict__ xz,
                                                   const float* __restrict__ cw,
                                                   const float* __restrict__ cb,
                                                   float* __restrict__ xc) {
  const size_t idx = (size_t)blockIdx.x * 256 + threadIdx.x;
  if (idx >= (size_t)ROWS * DI) return;
  const int d = (int)(idx % DI);
  const int r = (int)(idx / DI);
  const int b = r / SEQ, t = r % SEQ;
  float acc = cb[d];
#pragma unroll
  for (int k = 0; k < 4; ++k) {
    const int tt = t - 3 + k;
    if (tt >= 0)
      acc += xz[(size_t)(b * SEQ + tt) * (2 * DI) + d] * cw[d * 4 + k];
  }
  const float sig = 1.0f / (1.0f + expf(-acc));
  xc[idx] = acc * sig;
}

// ---------------- K4: x_ssm = xc @ w_x  (N=33 GEMV per row) ----------------
__global__ __launch_bounds__(256) void xssm_k(const float* __restrict__ xc,
                                              const float* __restrict__ w_x,
                                              float* __restrict__ ssm) {
  const int row = blockIdx.x;
  const float* xr = xc + (size_t)row * DI;
  float acc[33];
#pragma unroll
  for (int j = 0; j < 33; ++j) acc[j] = 0.f;
  for (int k = threadIdx.x; k < DI; k += 256) {
    const float xv = xr[k];
    const float* wr = w_x + (size_t)k * 33;
#pragma unroll
    for (int j = 0; j < 33; ++j) acc[j] += xv * wr[j];
  }
#pragma unroll
  for (int j = 0; j < 33; ++j)
#pragma unroll
    for (int m = 16; m >= 1; m >>= 1) acc[j] += __shfl_xor(acc[j], m, 32);
  __shared__ float red[8][33];
  const int wv = threadIdx.x >> 5, ln = threadIdx.x & 31;
  if (ln == 0)
#pragma unroll
    for (int j = 0; j < 33; ++j) red[wv][j] = acc[j];
  __syncthreads();
  if (threadIdx.x < 33) {
    float t = 0.f;
#pragma unroll
    for (int w = 0; w < 8; ++w) t += red[w][threadIdx.x];
    ssm[(size_t)row * 33 + threadIdx.x] = t;
  }
}

// ---------------- K5: selective scan (16 lanes = 16 states / channel) ------
__global__ __launch_bounds__(256) void scan_k(const float* __restrict__ ssm,
                                              const float* __restrict__ xc,
                                              const float* __restrict__ xz,
                                              const float* __restrict__ A_log,
                                              const float* __restrict__ w_dt,
                                              const float* __restrict__ b_dt,
                                              const float* __restrict__ Dp,
                                              float* __restrict__ yb) {
  const int grp = threadIdx.x >> 4;              // 16 channels per block
  const int s   = threadIdx.x & 15;              // state index
  const int ch  = blockIdx.x * 16 + grp;         // 0..B*DI-1
  const int b   = ch / DI, d = ch % DI;

  const float Av  = -expf(A_log[d * DS + s]);
  const float wdt = w_dt[d], bdt = b_dt[d], Dd = Dp[d];
  float h = 0.f;

  const float* ssmb = ssm + (size_t)b * SEQ * 33;
  const float* xcb  = xc + (size_t)b * SEQ * DI + d;
  const float* zb   = xz + (size_t)b * SEQ * (2 * DI) + DI + d;
  float* yo         = yb + (size_t)b * SEQ * DI + d;

  for (int t = 0; t < SEQ; ++t) {
    const float* sr = ssmb + (size_t)t * 33;
    const float Bv  = sr[s];
    const float Cv  = sr[16 + s];
    const float dtr = sr[32];
    const float xt  = xcb[(size_t)t * DI];

    const float u  = dtr * wdt + bdt;
    const float dt = (u > 20.0f) ? u : log1pf(expf(u));   // softplus

    h = expf(Av * dt) * h + (dt * xt) * Bv;

    float yp = h * Cv;                       // reduce across the 16 states
    yp += __shfl_xor(yp, 1, 32);
    yp += __shfl_xor(yp, 2, 32);
    yp += __shfl_xor(yp, 4, 32);
    yp += __shfl_xor(yp, 8, 32);

    if (s == 0) {
      const float zv  = zb[(size_t)t * (2 * DI)];
      const float y   = yp + xt * Dd;
      const float sig = 1.0f / (1.0f + expf(-zv));
      yo[(size_t)t * DI] = y * (zv * sig);   // y * silu(z)
    }
  }
}

// ---------------------------------------------------------------------------
extern "C" void kernel_launch(void* const* d_in, const int* in_sizes, int n_in,
                              void* d_out, int out_size, void* d_ws, size_t ws_size,
                              hipStream_t stream) {
  const float* x      = (const float*)d_in[0];
  const float* norm_w = (const float*)d_in[1];
  const float* w_in   = (const float*)d_in[2];
  const float* conv_w = (const float*)d_in[3];
  const float* conv_b = (const float*)d_in[4];
  const float* w_x    = (const float*)d_in[5];
  const float* w_dt   = (const float*)d_in[6];
  const float* b_dt   = (const float*)d_in[7];
  const float* A_log  = (const float*)d_in[8];
  const float* Dp     = (const float*)d_in[9];
  const float* w_out  = (const float*)d_in[10];
  float* out = (float*)d_out;

  // Workspace layout (floats): ~113.8 MB total.
  float* xn  = (float*)d_ws;                        // 4096*768
  float* xz  = xn  + (size_t)ROWS * DM;             // 4096*3072
  float* xc  = xz  + (size_t)ROWS * 2 * DI;         // 4096*1536
  float* ssm = xc  + (size_t)ROWS * DI;             // 4096*33
  float* yb  = ssm + (size_t)ROWS * 33;             // 4096*1536

  const size_t smem_bytes = (size_t)SMEM_FLOATS * sizeof(float);

  // K1: RMSNorm
  rmsnorm_k<<<ROWS, 256, 0, stream>>>(x, norm_w, xn);

  // K2: xz = xn @ w_in   (M=4096, N=3072, K=768)
  {
    dim3 g(2 * DI / BN, ROWS / BM);
    gemm_wmma_f32<<<g, 256, smem_bytes, stream>>>(xn, w_in, nullptr, xz, ROWS, 2 * DI, DM);
  }

  // K3: depthwise conv + SiLU
  {
    const size_t n = (size_t)ROWS * DI;
    conv_silu_k<<<(unsigned)((n + 255) / 256), 256, 0, stream>>>(xz, conv_w, conv_b, xc);
  }

  // K4: ssm = xc @ w_x (33 outputs per row)
  xssm_k<<<ROWS, 256, 0, stream>>>(xc, w_x, ssm);

  // K5: selective scan + skip + gate
  scan_k<<<(BSZ * DI) / 16, 256, 0, stream>>>(ssm, xc, xz, A_log, w_dt, b_dt, Dp, yb);

  // K6: out = residual + yb @ w_out   (M=4096, N=768, K=1536)
  {
    dim3 g(DM / BN, ROWS / BM);
    gemm_wmma_f32<<<g, 256, smem_bytes, stream>>>(yb, w_out, x, out, ROWS, DM, DI);
  }
}